// GCNConv_75917841924787
// MI455X (gfx1250) — compile-verified
//
#include <hip/hip_runtime.h>

#define IN_CH  256
#define OUT_CH 64

typedef __attribute__((ext_vector_type(2))) float v2f;
typedef __attribute__((ext_vector_type(8))) float v8f;

// ---------------------------------------------------------------------------
// K0: deg = 1.0 (degree includes self), agg = 0
// ---------------------------------------------------------------------------
__global__ __launch_bounds__(256) void gcn_init(float* __restrict__ deg,
                                                float* __restrict__ agg,
                                                int n_nodes) {
  unsigned i = blockIdx.x * blockDim.x + threadIdx.x;
  unsigned total = (unsigned)n_nodes * OUT_CH;
  if (i < total) agg[i] = 0.0f;
  if (i < (unsigned)n_nodes) deg[i] = 1.0f;
}

// ---------------------------------------------------------------------------
// K1: deg[src[e]] += 1 for every edge
// ---------------------------------------------------------------------------
__global__ __launch_bounds__(256) void gcn_degree(const int* __restrict__ src,
                                                  float* __restrict__ deg,
                                                  int n_edges) {
  unsigned e = blockIdx.x * blockDim.x + threadIdx.x;
  if (e < (unsigned)n_edges) {
    atomicAdd(&deg[src[e]], 1.0f);
  }
}

// ---------------------------------------------------------------------------
// K2: norm = rsqrt(deg), in place
// ---------------------------------------------------------------------------
__global__ __launch_bounds__(256) void gcn_rsqrt(float* __restrict__ deg,
                                                 int n_nodes) {
  unsigned i = blockIdx.x * blockDim.x + threadIdx.x;
  if (i < (unsigned)n_nodes) deg[i] = rsqrtf(deg[i]);
}

// ---------------------------------------------------------------------------
// K3: h = x @ theta  via V_WMMA_F32_16X16X4_F32 (exact fp32 matrix pipe)
//   block = 128 threads = 4 wave32; wave w computes output cols [16w, 16w+16)
//   for the 16-row tile  [16*blockIdx.x, 16*blockIdx.x+16).
//   x-tile (16x256 fp32 = 16KB) staged in LDS; theta (64KB, hot) via caches.
// ---------------------------------------------------------------------------
__global__ __launch_bounds__(128) void gcn_gemm(const float* __restrict__ x,
                                                const float* __restrict__ theta,
                                                float* __restrict__ h,
                                                int n_nodes) {
  __shared__ float sX[16 * IN_CH];   // 16 KB

  const int tid  = threadIdx.x;       // 0..127
  const int wave = tid >> 5;          // 0..3  -> N tile
  const int lane = tid & 31;
  const int row0 = blockIdx.x * 16;
  const bool full_tile = (row0 + 16 <= n_nodes);

  // Cooperative load of the 16xIN_CH x-tile as float4 (coalesced b128 loads).
  {
    const float4* xg = (const float4*)(x + (size_t)row0 * IN_CH);
    float4* xs = (float4*)sX;
    const int n4 = 16 * IN_CH / 4;          // 1024 float4
    const int f4row = IN_CH / 4;            // 64 float4 per row
    if (full_tile) {
      for (int i = tid; i < n4; i += 128) xs[i] = xg[i];
    } else {
      for (int i = tid; i < n4; i += 128) {
        int r = i / f4row;
        float4 v = make_float4(0.f, 0.f, 0.f, 0.f);
        if (row0 + r < n_nodes) v = xg[i];
        xs[i] = v;
      }
    }
  }
  __syncthreads();

  // WMMA fragment indexing (ISA 7.12.2, 32-bit 16x4 A / 16x16 C layouts)
  const int m    = lane & 15;                  // A row for this lane
  const int kf   = (lane < 16) ? 0 : 2;        // K sub-offset for this half-wave
  const int ncol = (wave << 4) + (lane & 15);  // B/C column for this lane

  v8f acc = {};
  const float* srcA = sX + m * IN_CH + kf;
  const float* srcB = theta + (size_t)kf * OUT_CH + ncol;

  #pragma unroll 4
  for (int k = 0; k < IN_CH; k += 4) {
    v2f a, b;
    a[0] = srcA[k];                       // A[m][k+kf]     (ds_load)
    a[1] = srcA[k + 1];                   // A[m][k+kf+1]
    b[0] = srcB[(size_t)k * OUT_CH];          // theta[k+kf][ncol]
    b[1] = srcB[(size_t)(k + 1) * OUT_CH];    // theta[k+kf+1][ncol]
    acc = __builtin_amdgcn_wmma_f32_16x16x4_f32(
        /*neg_a=*/false, a, /*neg_b=*/false, b,
        /*c_mod=*/(short)0, acc, /*reuse_a=*/false, /*reuse_b=*/false);
  }

  // Store C: VGPR r holds row (r + 0/8 by half-wave), col = lane&15 per tile.
  const int mbase = (lane < 16) ? 0 : 8;
  float* hp = h + (size_t)(row0 + mbase) * OUT_CH + ncol;
  if (full_tile) {
    #pragma unroll
    for (int r = 0; r < 8; ++r) hp[(size_t)r * OUT_CH] = acc[r];
  } else {
    #pragma unroll
    for (int r = 0; r < 8; ++r) {
      if (row0 + mbase + r < n_nodes) hp[(size_t)r * OUT_CH] = acc[r];
    }
  }
}

// ---------------------------------------------------------------------------
// K4: agg[src[e]][c] += norm[dst[e]] * h[dst[e]][c]
//   one thread per (edge, 4-channel group): indices + norm loaded once per
//   quad, h gathered as float4 (b128), 4x global_atomic_add_f32 (irreducible).
// ---------------------------------------------------------------------------
__global__ __launch_bounds__(256) void gcn_scatter(const int* __restrict__ src,
                                                   const int* __restrict__ dst,
                                                   const float* __restrict__ norm,
                                                   const float* __restrict__ h,
                                                   float* __restrict__ agg,
                                                   int n_edges) {
  unsigned idx = blockIdx.x * blockDim.x + threadIdx.x;
  unsigned e = idx >> 4;            // 16 threads per edge
  unsigned c = (idx & 15) << 2;     // channel group of 4
  if (e < (unsigned)n_edges) {
    int s = src[e];
    int d = dst[e];
    float nv = norm[d];
    float4 hv = *(const float4*)(h + (size_t)d * OUT_CH + c);
    float* ap = agg + (size_t)s * OUT_CH + c;
    atomicAdd(ap + 0, nv * hv.x);
    atomicAdd(ap + 1, nv * hv.y);
    atomicAdd(ap + 2, nv * hv.z);
    atomicAdd(ap + 3, nv * hv.w);
  }
}

// ---------------------------------------------------------------------------
// K5: out = norm * (norm*h + agg), float4 per thread
// ---------------------------------------------------------------------------
__global__ __launch_bounds__(256) void gcn_finalize(const float* __restrict__ norm,
                                                    const float* __restrict__ h,
                                                    const float* __restrict__ agg,
                                                    float* __restrict__ out,
                                                    int n_nodes) {
  unsigned i = blockIdx.x * blockDim.x + threadIdx.x;   // float4 index
  unsigned total4 = (unsigned)n_nodes * (OUT_CH / 4);
  if (i < total4) {
    unsigned node = i >> 4;          // 16 float4 per node
    float nv = norm[node];
    float4 hv = ((const float4*)h)[i];
    float4 av = ((const float4*)agg)[i];
    float4 o;
    o.x = nv * (nv * hv.x + av.x);
    o.y = nv * (nv * hv.y + av.y);
    o.z = nv * (nv * hv.z + av.z);
    o.w = nv * (nv * hv.w + av.w);
    ((float4*)out)[i] = o;
  }
}

// ---------------------------------------------------------------------------
extern "C" void kernel_launch(void* const* d_in, const int* in_sizes, int n_in,
                              void* d_out, int out_size, void* d_ws, size_t ws_size,
                              hipStream_t stream) {
  const float* x     = (const float*)d_in[0];   // (n_nodes, 256)
  const float* theta = (const float*)d_in[1];   // (256, 64)
  const int*   ei    = (const int*)d_in[2];     // (2, n_edges)

  const int n_nodes = in_sizes[0] / IN_CH;
  const int n_edges = in_sizes[2] / 2;
  const int* src = ei;                // edge_index[0]
  const int* dst = ei + n_edges;      // edge_index[1]

  // Workspace layout: [deg/norm : n_nodes] [h : n_nodes*64] [agg : n_nodes*64]
  float* deg = (float*)d_ws;
  float* h   = deg + n_nodes;
  float* agg = h + (size_t)n_nodes * OUT_CH;

  float* out = (float*)d_out;

  const unsigned nc_total = (unsigned)n_nodes * OUT_CH;

  gcn_init<<<(nc_total + 255) / 256, 256, 0, stream>>>(deg, agg, n_nodes);
  gcn_degree<<<((unsigned)n_edges + 255) / 256, 256, 0, stream>>>(src, deg, n_edges);
  gcn_rsqrt<<<((unsigned)n_nodes + 255) / 256, 256, 0, stream>>>(deg, n_nodes);
  gcn_gemm<<<(n_nodes + 15) / 16, 128, 0, stream>>>(x, theta, h, n_nodes);
  {
    unsigned work = (unsigned)n_edges * 16u;   // 16 threads (4ch each) per edge
    gcn_scatter<<<(work + 255) / 256, 256, 0, stream>>>(src, dst, deg, h, agg, n_edges);
  }
  {
    unsigned work4 = (unsigned)n_nodes * (OUT_CH / 4);
    gcn_finalize<<<(work4 + 255) / 256, 256, 0, stream>>>(deg, h, agg, out, n_nodes);
  }
}